// Pooling_5866925326754
// MI455X (gfx1250) — compile-verified
//
#include <hip/hip_runtime.h>
#include <hip/hip_bf16.h>

// Pooling over pairs of 3x3 matrices: keep the one with larger rotation angle.
// acos(clip((tr-1)/2)) is strictly decreasing in the clipped half-trace, so
// "larger angle" == "smaller clipped half-trace"; ties -> index 0 (matches
// jnp.argmax first-max semantics).
//
// Memory-bound (453 MB @ 23.3 TB/s ~= 19.4 us). CDNA5 path: async DMA into LDS
// (ASYNCcnt-tracked, double-buffered, GVS addressing + shared IOFFSET trick),
// wave32 512B-coalesced b128 transfers, NT cache hints both directions.

typedef float v4f __attribute__((ext_vector_type(4)));

#define NMAT   256          // num axis length
#define FDIM   128          // trailing frame axis
#define NCOMP  9            // 3x3 components
#define NB     8            // num values per tile (4 groups)
#define GPT    (NB / 2)     // groups per tile
#define NTHR   128          // threads per block (4 waves)
#define TILES_PER_N   (NMAT / NB)            // 32
#define TOTAL_TILES   (256 * TILES_PER_N)    // 8192
#define GRID_BLOCKS   2048                   // 4 tiles per block
#define CHUNK_GBYTES  (NMAT * FDIM * 4)      // 131072 B between components
#define CHUNK_LBYTES  (NB * FDIM * 4)        // 4096 B between components in LDS

__global__ __launch_bounds__(NTHR)
void rotpool_async_kernel(const float* __restrict__ x, float* __restrict__ out) {
    // Double-buffered staging: 2 x 9 comps x 8 nums x 128 f floats = 72 KB
    __shared__ __align__(16) float lds[2][NCOMP * NB * FDIM];

    const int t = threadIdx.x;

    // Per-lane invariants for the DMA phase.
    const unsigned voff0 = (unsigned)(t * 16);   // byte offset of lane's float4
    const unsigned lbase[2] = {
        (unsigned)(uintptr_t)&lds[0][t * 4],     // low 32 bits == LDS byte addr
        (unsigned)(uintptr_t)&lds[1][t * 4],
    };

    // ---- async DMA of one tile's input into LDS buffer `buf` ----
    // x flat index: ((n*9 + c)*256 + m)*128 + f. Per-component chunk of
    // NB*128 floats (2048 B per lane-pair step) is contiguous. The async
    // IOFFSET is added to BOTH the global and LDS addresses, and the two
    // loads per thread within a chunk differ by +2048 B on both sides, so
    // each chunk needs one address setup and two issues.
    auto issue_tile = [&](int tile, int buf) {
        const int n  = tile >> 5;          // tile / TILES_PER_N
        const int mb = tile & 31;          // num-block index; num0 = mb*NB
        const unsigned long long sbase = (unsigned long long)(uintptr_t)
            (x + (size_t)n * (NCOMP * NMAT * FDIM) + (size_t)mb * (NB * FDIM));
#pragma unroll
        for (int c = 0; c < NCOMP; ++c) {
            const unsigned vo = voff0 + (unsigned)(c * CHUNK_GBYTES);
            const unsigned lo = lbase[buf] + (unsigned)(c * CHUNK_LBYTES);
            asm volatile(
                "global_load_async_to_lds_b128 %0, %1, %2 th:TH_LOAD_NT"
                :: "v"(lo), "v"(vo), "s"(sbase) : "memory");
            asm volatile(
                "global_load_async_to_lds_b128 %0, %1, %2 offset:2048 th:TH_LOAD_NT"
                :: "v"(lo), "v"(vo), "s"(sbase) : "memory");
        }
    };

    // ---- select + store one tile from LDS buffer `buf` ----
    auto compute_tile = [&](int tile, int buf) {
        const int n  = tile >> 5;
        const int mb = tile & 31;
        const int gl = t >> 5;             // group-in-tile 0..3 (one per wave)
        const int f4 = t & 31;             // lane -> 4 consecutive f
        const int fo = f4 * 4;
        const int m0 = gl * 2;             // candidate 0 (num even)
        const int m1 = m0 + 1;             // candidate 1 (num odd)
        const float* L = lds[buf];

        auto LD = [&](int c, int m) -> v4f {
            return *(const v4f*)&L[(c * NB + m) * FDIM + fo];   // ds_load_b128
        };

        // traces from diagonal components c = 0 (0,0), 4 (1,1), 8 (2,2)
        const v4f t0 = LD(0, m0) + LD(4, m0) + LD(8, m0);
        const v4f t1 = LD(0, m1) + LD(4, m1) + LD(8, m1);

        bool pick1[4];
#pragma unroll
        for (int i = 0; i < 4; ++i) {
            const float c0 = fminf(1.0f, fmaxf(-1.0f, (t0[i] - 1.0f) * 0.5f));
            const float c1 = fminf(1.0f, fmaxf(-1.0f, (t1[i] - 1.0f) * 0.5f));
            pick1[i] = (c1 < c0);          // strictly larger angle -> index 1
        }

        const int g = mb * GPT + gl;       // global group index 0..127
#pragma unroll
        for (int c = 0; c < NCOMP; ++c) {
            const v4f A = LD(c, m0);
            const v4f B = LD(c, m1);
            v4f R;
#pragma unroll
            for (int i = 0; i < 4; ++i) R[i] = pick1[i] ? B[i] : A[i];
            // out flat index: ((n*9 + c)*128 + g)*128 + f  (streaming, NT)
            float* op = out + (((size_t)(n * NCOMP + c)) * 128 + g) * FDIM + fo;
            __builtin_nontemporal_store(R, (v4f*)op);
        }
    };

    // ---- double-buffered pipeline over this block's tiles ----
    int tile = blockIdx.x;
    int buf  = 0;
    if (tile < TOTAL_TILES) issue_tile(tile, buf);
    while (tile < TOTAL_TILES) {
        const int next = tile + GRID_BLOCKS;
        if (next < TOTAL_TILES) {
            issue_tile(next, buf ^ 1);
            // 36 outstanding; retire the oldest 18 (current buffer, in order)
            asm volatile("s_wait_asynccnt 0x12" ::: "memory");
        } else {
            asm volatile("s_wait_asynccnt 0x0" ::: "memory");
        }
        __syncthreads();                   // all waves' DMA for `buf` landed
        compute_tile(tile, buf);
        __syncthreads();                   // done reading `buf` before refill
        buf ^= 1;
        tile = next;
    }
}

extern "C" void kernel_launch(void* const* d_in, const int* in_sizes, int n_in,
                              void* d_out, int out_size, void* d_ws, size_t ws_size,
                              hipStream_t stream) {
    (void)in_sizes; (void)n_in; (void)out_size; (void)d_ws; (void)ws_size;
    const float* x = (const float*)d_in[0];
    float* out = (float*)d_out;
    hipLaunchKernelGGL(rotpool_async_kernel,
                       dim3(GRID_BLOCKS), dim3(NTHR), 0, stream,
                       x, out);
}